// WordEmbed_7971459302123
// MI455X (gfx1250) — compile-verified
//
#include <hip/hip_runtime.h>

// Embedding gather: out[row,:] = emb[ids[row],:]
//   ids: [2,2048] int32  -> 4096 tokens
//   emb: [32000,1024] f32
//   out: [4096,1024] f32
// Pure HBM-bound data movement (~34MB => ~1.4us at 23.3 TB/s). MI455X path:
// TDM gather-mode DMA (tensor_load_to_lds with 16-bit row indices) + TDM
// contiguous store; payload never enters VGPRs; synchronized via TENSORcnt.
// 512 single-wave workgroups, 32KB LDS slab each, pipelined across WGPs.

#define VOCAB        32000
#define FEAT         1024
#define TOKENS_TOTAL (2 * 2048)
#define ROWS_PER_WG  8            // 8 rows * 4KB = 32KB LDS per workgroup

typedef __attribute__((ext_vector_type(4))) unsigned int v4u;
typedef __attribute__((ext_vector_type(8))) int          v8i;
typedef __attribute__((ext_vector_type(4))) int          v4i;

#if defined(__has_builtin)
#if __has_builtin(__builtin_amdgcn_tensor_load_to_lds) && \
    __has_builtin(__builtin_amdgcn_tensor_store_from_lds) && \
    __has_builtin(__builtin_amdgcn_s_wait_tensorcnt)
#define USE_TDM 1
#endif
#endif
#ifndef USE_TDM
#define USE_TDM 0
#endif

#if USE_TDM
__global__ __launch_bounds__(32) void embed_gather_tdm(
    const int* __restrict__ ids, const float* __restrict__ emb,
    float* __restrict__ out) {
  __shared__ float tile[ROWS_PER_WG * FEAT];  // 32 KB staging slab

  const unsigned wg   = blockIdx.x;
  const unsigned row0 = wg * ROWS_PER_WG;

  // This workgroup's 8 token ids (uniform -> one s_load_b256), packed as
  // 16-bit row indices, two per dword, for D# gather-mode group 2.
  unsigned packed[ROWS_PER_WG / 2];
#pragma unroll
  for (int k = 0; k < ROWS_PER_WG / 2; ++k) {
    unsigned i0 = (unsigned)ids[row0 + 2 * k];
    unsigned i1 = (unsigned)ids[row0 + 2 * k + 1];
    packed[k] = (i0 & 0xFFFFu) | (i1 << 16);
  }

  const unsigned lds_base = (unsigned)(uintptr_t)(&tile[0]);
  const unsigned long long embA = (unsigned long long)(uintptr_t)emb;
  const unsigned long long outA = (unsigned long long)(uintptr_t)out +
      (unsigned long long)row0 * FEAT * sizeof(float);

  // ---------------- Load descriptor: gather mode, 16-bit indices -----------
  v4u g0l;
  g0l.x = 1u            // count = 1 (valid descriptor)
        | (0u << 30)    // gather_index_size = 0 (16-bit indices)
        | (1u << 31);   // gather_mode = 1
  g0l.y = lds_base;                                       // lds_addr
  g0l.z = (unsigned)embA;                                 // global_addr[31:0]
  g0l.w = ((unsigned)(embA >> 32) & 0x01FFFFFFu)          // global_addr[56:32]
        | (2u << 30);                                     // type = 2 ("image")

  v8i g1l;
  g1l[0] = (int)(2u << 16);                 // workgroup_mask=0, data_size=2 (4B)
  g1l[1] = (int)(((unsigned)FEAT & 0xFFFFu) << 16);   // tensor_dim0 lo16 @ [63:48]
  g1l[2] = (int)(((unsigned)VOCAB & 0xFFFFu) << 16)   // tensor_dim1 lo16 @ [95:80]
         | (int)((unsigned)FEAT >> 16);               // tensor_dim0 hi16 @ [79:64]
  g1l[3] = (int)(((unsigned)FEAT & 0xFFFFu) << 16)    // tile_dim0 @ [127:112]
         | (int)((unsigned)VOCAB >> 16);              // tensor_dim1 hi16 @ [111:96]
  g1l[4] = ROWS_PER_WG;                     // tile_dim1 = #valid gather indices
  g1l[5] = FEAT;                            // tensor_dim0_stride lo32 (elements)
  g1l[6] = 0;                               // stride hi16 / dim1_stride (ignored)
  g1l[7] = 0;

  v4i g2l;                                  // row_index_0..7 (2 per dword)
  g2l[0] = (int)packed[0];
  g2l[1] = (int)packed[1];
  g2l[2] = (int)packed[2];
  g2l[3] = (int)packed[3];
  v4i g3z = {0, 0, 0, 0};                   // row_index_8..15 unused
  v8i gxz = {0, 0, 0, 0, 0, 0, 0, 0};       // extra descriptor group (disabled)

  // clang-23 / therock-10 form: (g0, g1, g2, g3, extra, cpol)
  __builtin_amdgcn_tensor_load_to_lds(g0l, g1l, g2l, g3z, gxz, 0);
  __builtin_amdgcn_s_wait_tensorcnt(0);     // LDS slab now holds gathered rows

  // ------------- Store descriptor: contiguous 8x1024 tile, normal mode -----
  v4u g0s;
  g0s.x = 1u;                               // count=1, gather_mode=0
  g0s.y = lds_base;
  g0s.z = (unsigned)outA;
  g0s.w = ((unsigned)(outA >> 32) & 0x01FFFFFFu) | (2u << 30);

  v8i g1s;
  g1s[0] = (int)(2u << 16);                             // data_size = 4B
  g1s[1] = (int)(((unsigned)FEAT & 0xFFFFu) << 16);     // tensor_dim0 = 1024
  g1s[2] = (int)(((unsigned)ROWS_PER_WG & 0xFFFFu) << 16); // tensor_dim1 = 8
  g1s[3] = (int)(((unsigned)FEAT & 0xFFFFu) << 16);     // tile_dim0 = 1024
  g1s[4] = ROWS_PER_WG;                                 // tile_dim1 = 8
  g1s[5] = FEAT;                                        // tensor_dim0_stride
  g1s[6] = 0;
  g1s[7] = 0;

  v4i g2z = {0, 0, 0, 0};
  __builtin_amdgcn_tensor_store_from_lds(g0s, g1s, g2z, g3z, gxz, 0);
  // s_endpgm performs an implicit wait-idle, draining TENSORcnt.
}

#else  // !USE_TDM
// Fallback (only compiled if TDM builtins are unavailable): one block per
// token row, one float4 per lane — b128 HBM-bound memcpy-gather.
__global__ __launch_bounds__(256) void embed_gather_vec(
    const int* __restrict__ ids, const float* __restrict__ emb,
    float* __restrict__ out) {
  const unsigned row = blockIdx.x;
  const int id = ids[row];  // uniform -> scalar load
  const float4* __restrict__ src =
      (const float4*)(emb + (size_t)id * FEAT);
  float4* __restrict__ dst = (float4*)(out + (size_t)row * FEAT);
  dst[threadIdx.x] = src[threadIdx.x];
}
#endif  // USE_TDM

extern "C" void kernel_launch(void* const* d_in, const int* in_sizes, int n_in,
                              void* d_out, int out_size, void* d_ws,
                              size_t ws_size, hipStream_t stream) {
  const int*   ids = (const int*)d_in[0];
  const float* emb = (const float*)d_in[1];
  float*       out = (float*)d_out;

#if USE_TDM
  embed_gather_tdm<<<TOKENS_TOTAL / ROWS_PER_WG, 32, 0, stream>>>(ids, emb, out);
#else
  embed_gather_vec<<<TOKENS_TOTAL, 256, 0, stream>>>(ids, emb, out);
#endif
}